// TrackEncoder_36490042147414
// MI455X (gfx1250) — compile-verified
//
#include <hip/hip_runtime.h>
#include <math.h>

// Problem dims (fixed by the reference)
#define SEQ     8          // B*T
#define LSEQ    4096       // sequence length
#define MROWS   (SEQ*LSEQ) // 32768 rows
#define DTRACK  64
#define DINNER  128
#define NSTATE  16
#define RRANK   4
#define XDCOLS  36         // RRANK + 2*NSTATE

// Chunked scan config: 32 chunks of 128 steps -> 16384 waves in pass1/pass3
#define NCHUNK  32
#define CLEN    (LSEQ / NCHUNK)          // 128
#define NCHAINS (SEQ * DINNER * NSTATE)  // 16384

typedef float v2f __attribute__((ext_vector_type(2)));
typedef float v8f __attribute__((ext_vector_type(8)));

__device__ __forceinline__ float silu_f(float x) { return x / (1.f + __expf(-x)); }

// ---------------------------------------------------------------------------
// WMMA f32 GEMM:  C[M,N] = A[M,K] @ B[K,N] + bias[N] (+ resid[M,N])
// Block = 256 threads = 8 waves covering 128 rows x (16*NT) cols.
// The B panel (KDIM x 16*NT) is staged into LDS ONCE per block with
// gfx1250 async global->LDS copies (ASYNCcnt), then shared by all 8 waves.
// Math uses V_WMMA_F32_16X16X4_F32 (full fp32, matches reference precision).
// ---------------------------------------------------------------------------
template<int KDIM, int NT>
__global__ __launch_bounds__(256)
void gemm_wmma_f32(const float* __restrict__ A, const float* __restrict__ B,
                   const float* __restrict__ bias, const float* __restrict__ resid,
                   float* __restrict__ C, int N) {
  constexpr int NP = 16 * NT;              // panel width
  __shared__ float sB[KDIM * NP];

  const int tid = threadIdx.x;
  const int n0b = blockIdx.x * NP;

  // ---- async stage B[0:KDIM, n0b:n0b+NP) into LDS ----
  unsigned ldsBase = (unsigned)(size_t)(void*)sB;   // LDS byte offset (addr[31:0])
  for (int i = tid; i < KDIM * NP; i += 256) {
    int k = i / NP;
    int c = i - k * NP;
    const float* gp = B + (size_t)k * N + n0b + c;
    unsigned loff = ldsBase + (unsigned)i * 4u;
    asm volatile("global_load_async_to_lds_b32 %0, %1, off"
                 :: "v"(loff), "v"(gp) : "memory");
  }
  asm volatile("s_wait_asynccnt 0x0" ::: "memory");
  __syncthreads();

  // ---- WMMA compute ----
  const int lane = tid & 31;
  const int half = lane >> 4;
  const int idx  = lane & 15;
  const int m0   = blockIdx.y * 128 + (tid >> 5) * 16;

  v8f zero = {};
  v8f acc[NT];
#pragma unroll
  for (int t = 0; t < NT; ++t) acc[t] = zero;

  const float* arow = A + (size_t)(m0 + idx) * KDIM + 2 * half;
  for (int k0 = 0; k0 < KDIM; k0 += 4) {
    v2f a; a.x = arow[k0]; a.y = arow[k0 + 1];       // A[m][k0+2h .. k0+2h+1]
    const float* brow = sB + (k0 + 2 * half) * NP;   // LDS rows k0+2h, k0+2h+1
#pragma unroll
    for (int t = 0; t < NT; ++t) {
      int c = t * 16 + idx;
      v2f b; b.x = brow[c]; b.y = brow[NP + c];
      acc[t] = __builtin_amdgcn_wmma_f32_16x16x4_f32(
          /*neg_a=*/false, a, /*neg_b=*/false, b,
          /*c_mod=*/(short)0, acc[t], /*reuse_a=*/false, /*reuse_b=*/false);
    }
  }
#pragma unroll
  for (int t = 0; t < NT; ++t) {
#pragma unroll
    for (int v = 0; v < 8; ++v) {
      int row = m0 + v + 8 * half;
      int col = n0b + t * 16 + idx;
      float val = acc[t][v] + bias[col];
      if (resid) val += resid[(size_t)row * N + col];
      C[(size_t)row * N + col] = val;
    }
  }
}

// ---------------------------------------------------------------------------
// Input projection: tracks (8,6,4096) -> x0 (32768, 64)
// ---------------------------------------------------------------------------
__global__ void inproj_kernel(const float* __restrict__ tracks,
                              const float* __restrict__ W,
                              const float* __restrict__ b,
                              float* __restrict__ x0) {
  size_t gid = (size_t)blockIdx.x * blockDim.x + threadIdx.x;
  if (gid >= (size_t)MROWS * DTRACK) return;
  int d = gid & (DTRACK - 1);
  size_t row = gid >> 6;
  int s = (int)(row / LSEQ);
  int l = (int)(row % LSEQ);
  float acc = b[d];
#pragma unroll
  for (int ch = 0; ch < 6; ++ch)
    acc += tracks[((size_t)s * 6 + ch) * LSEQ + l] * W[ch * DTRACK + d];
  x0[gid] = acc;
}

// ---------------------------------------------------------------------------
// RMSNorm over last dim (64). One wave per row, butterfly reduce.
// ---------------------------------------------------------------------------
__global__ __launch_bounds__(256)
void rmsnorm_kernel(const float* __restrict__ x, const float* __restrict__ scale,
                    float* __restrict__ xn) {
  int row  = blockIdx.x * 8 + (threadIdx.x >> 5);
  int lane = threadIdx.x & 31;
  const float* xr = x + (size_t)row * DTRACK + lane * 2;
  float a0 = xr[0], a1 = xr[1];
  float ss = a0 * a0 + a1 * a1;
  ss += __shfl_xor(ss, 16, 32);
  ss += __shfl_xor(ss, 8, 32);
  ss += __shfl_xor(ss, 4, 32);
  ss += __shfl_xor(ss, 2, 32);
  ss += __shfl_xor(ss, 1, 32);
  float r = rsqrtf(ss * (1.f / DTRACK) + 1e-6f);
  float* o = xn + (size_t)row * DTRACK + lane * 2;
  o[0] = a0 * r * scale[lane * 2];
  o[1] = a1 * r * scale[lane * 2 + 1];
}

// ---------------------------------------------------------------------------
// Depthwise causal conv (K=4) + SiLU, with optional time flip.
// Reads xi = xz columns [0,128); xz row stride is 256.
// ---------------------------------------------------------------------------
__global__ void conv_silu_kernel(const float* __restrict__ xz,
                                 const float* __restrict__ W,
                                 const float* __restrict__ b,
                                 float* __restrict__ u, int reverse) {
  size_t gid = (size_t)blockIdx.x * blockDim.x + threadIdx.x;
  if (gid >= (size_t)MROWS * DINNER) return;
  int d = gid & (DINNER - 1);
  size_t r = gid >> 7;
  int s = (int)(r / LSEQ);
  int l = (int)(r % LSEQ);
  float acc = b[d];
#pragma unroll
  for (int k = 0; k < 4; ++k) {
    int pos = l - 3 + k;
    if (pos >= 0) {
      int sl = reverse ? (LSEQ - 1 - pos) : pos;
      acc += W[k * DINNER + d] * xz[((size_t)s * LSEQ + sl) * (2 * DINNER) + d];
    }
  }
  u[gid] = silu_f(acc);
}

// ---------------------------------------------------------------------------
// xd = u @ x_proj_W (128 -> 36). Ragged N, keep on VALU.
// ---------------------------------------------------------------------------
__global__ void xproj_kernel(const float* __restrict__ u,
                             const float* __restrict__ Wx,
                             float* __restrict__ xd) {
  size_t gid = (size_t)blockIdx.x * blockDim.x + threadIdx.x;
  if (gid >= (size_t)MROWS * XDCOLS) return;
  int j = (int)(gid % XDCOLS);
  size_t row = gid / XDCOLS;
  const float* ur = u + row * DINNER;
  float acc = 0.f;
#pragma unroll 8
  for (int i = 0; i < DINNER; ++i) acc += ur[i] * Wx[i * XDCOLS + j];
  xd[gid] = acc;
}

// ---------------------------------------------------------------------------
// dt = softplus(dt_low @ dt_proj_W + dt_proj_b)
// ---------------------------------------------------------------------------
__global__ void dtproj_kernel(const float* __restrict__ xd,
                              const float* __restrict__ Wdt,
                              const float* __restrict__ bdt,
                              float* __restrict__ dt) {
  size_t gid = (size_t)blockIdx.x * blockDim.x + threadIdx.x;
  if (gid >= (size_t)MROWS * DINNER) return;
  int d = gid & (DINNER - 1);
  size_t row = gid >> 7;
  float acc = bdt[d];
#pragma unroll
  for (int r = 0; r < RRANK; ++r)
    acc += xd[row * XDCOLS + r] * Wdt[r * DINNER + d];
  dt[gid] = (acc > 20.f) ? acc : log1pf(__expf(acc));
}

// ---------------------------------------------------------------------------
// Chunked selective scan, pass 1: each wave owns (seq s, channel pair, chunk);
// lanes [0,16) = state dims n for channel d0, lanes [16,32) for d0+1.
// Computes per-chunk (prod a, local scan final) with h_init = 0.
// ---------------------------------------------------------------------------
__global__ __launch_bounds__(256)
void scan_pass1_kernel(const float* __restrict__ dt, const float* __restrict__ u,
                       const float* __restrict__ xd, const float* __restrict__ A_log,
                       float* __restrict__ Pbuf, float* __restrict__ Sbuf) {
  int w    = blockIdx.x * (blockDim.x >> 5) + (threadIdx.x >> 5);
  int lane = threadIdx.x & 31;
  int n    = lane & 15;
  int half = lane >> 4;
  int chunk = w & (NCHUNK - 1);
  int rest  = w >> 5;                   // NCHUNK == 32
  int s     = rest >> 6;                // [0,8)
  int d     = ((rest & 63) << 1) + half;// [0,128)

  float Acoef = -__expf(A_log[d * NSTATE + n]);
  float P = 1.f, S = 0.f;
  int l0 = chunk * CLEN;
  for (int l = l0; l < l0 + CLEN; ++l) {
    size_t row = (size_t)s * LSEQ + l;
    float dtv = dt[row * DINNER + d];
    float uv  = u [row * DINNER + d];
    float Bv  = xd[row * XDCOLS + RRANK + n];
    __builtin_prefetch(&dt[(row + 1) * DINNER + d], 0, 0);
    float a = __expf(dtv * Acoef);
    P *= a;
    S = a * S + dtv * uv * Bv;
  }
  size_t chain = ((size_t)s * DINNER + d) * NSTATE + n;
  Pbuf[chain * NCHUNK + chunk] = P;
  Sbuf[chain * NCHUNK + chunk] = S;
}

// ---------------------------------------------------------------------------
// Pass 2: per chain, sequentially combine the 32 chunk summaries to get the
// h entering each chunk. 16384 threads, 32 steps each.
// ---------------------------------------------------------------------------
__global__ __launch_bounds__(256)
void scan_combine_kernel(const float* __restrict__ Pbuf, const float* __restrict__ Sbuf,
                         float* __restrict__ h0buf) {
  size_t chain = (size_t)blockIdx.x * blockDim.x + threadIdx.x;
  if (chain >= NCHAINS) return;
  float h = 0.f;
#pragma unroll 4
  for (int c = 0; c < NCHUNK; ++c) {
    size_t idx = chain * NCHUNK + c;
    h0buf[idx] = h;
    h = Pbuf[idx] * h + Sbuf[idx];
  }
}

// ---------------------------------------------------------------------------
// Pass 3: re-scan each chunk from its known initial state, contract with Cm
// via 16-lane shfl-xor butterfly, add the D skip term, and write y with the
// direction flip folded into the output index (bwd accumulates onto fwd).
// ---------------------------------------------------------------------------
__global__ __launch_bounds__(256)
void scan_pass3_kernel(const float* __restrict__ dt, const float* __restrict__ u,
                       const float* __restrict__ xd, const float* __restrict__ A_log,
                       const float* __restrict__ Dp, const float* __restrict__ h0buf,
                       float* __restrict__ y, int reverse, int accumulate) {
  int w    = blockIdx.x * (blockDim.x >> 5) + (threadIdx.x >> 5);
  int lane = threadIdx.x & 31;
  int n    = lane & 15;
  int half = lane >> 4;
  int chunk = w & (NCHUNK - 1);
  int rest  = w >> 5;
  int s     = rest >> 6;
  int d     = ((rest & 63) << 1) + half;

  float Acoef = -__expf(A_log[d * NSTATE + n]);
  float Dd = Dp[d];
  size_t chain = ((size_t)s * DINNER + d) * NSTATE + n;
  float h = h0buf[chain * NCHUNK + chunk];

  int l0 = chunk * CLEN;
  for (int l = l0; l < l0 + CLEN; ++l) {
    size_t row = (size_t)s * LSEQ + l;
    float dtv = dt[row * DINNER + d];
    float uv  = u [row * DINNER + d];
    float Bv  = xd[row * XDCOLS + RRANK + n];
    float Cv  = xd[row * XDCOLS + RRANK + NSTATE + n];
    __builtin_prefetch(&dt[(row + 1) * DINNER + d], 0, 0);
    float a = __expf(dtv * Acoef);
    h = a * h + dtv * uv * Bv;
    float c = h * Cv;
    c += __shfl_xor(c, 8, 32);          // reduce within 16-lane half
    c += __shfl_xor(c, 4, 32);
    c += __shfl_xor(c, 2, 32);
    c += __shfl_xor(c, 1, 32);
    if (n == 0) {
      int lout = reverse ? (LSEQ - 1 - l) : l;
      size_t oidx = ((size_t)s * LSEQ + lout) * DINNER + d;
      float yv = c + uv * Dd;
      y[oidx] = accumulate ? (y[oidx] + yv) : yv;
    }
  }
}

// ---------------------------------------------------------------------------
// yg = y * silu(z);  z = xz columns [128,256)
// ---------------------------------------------------------------------------
__global__ void gate_kernel(const float* __restrict__ y,
                            const float* __restrict__ xz,
                            float* __restrict__ yg) {
  size_t gid = (size_t)blockIdx.x * blockDim.x + threadIdx.x;
  if (gid >= (size_t)MROWS * DINNER) return;
  int d = gid & (DINNER - 1);
  size_t row = gid >> 7;
  float z = xz[row * (2 * DINNER) + DINNER + d];
  yg[gid] = y[gid] * silu_f(z);
}

// ---------------------------------------------------------------------------
// Host orchestration
// Input order (setup_inputs dict order): tracks, in_proj_W, in_proj_b,
// then per layer (19 tensors): norm_scale, in_W, in_b,
//   fwd{conv_W,conv_b,x_proj_W,dt_proj_W,dt_proj_b,A_log,D},
//   bwd{...}, out_W, out_b
// ---------------------------------------------------------------------------
extern "C" void kernel_launch(void* const* d_in, const int* in_sizes, int n_in,
                              void* d_out, int out_size, void* d_ws, size_t ws_size,
                              hipStream_t stream) {
  (void)in_sizes; (void)n_in; (void)out_size; (void)ws_size;
  const float* tracks = (const float*)d_in[0];
  const float* ipW    = (const float*)d_in[1];
  const float* ipB    = (const float*)d_in[2];

  float* ws = (float*)d_ws;
  size_t off = 0;
  float* x0 = ws + off; off += (size_t)MROWS * DTRACK;
  float* x1 = ws + off; off += (size_t)MROWS * DTRACK;
  float* xn = ws + off; off += (size_t)MROWS * DTRACK;
  float* xz = ws + off; off += (size_t)MROWS * 2 * DINNER;
  float* u  = ws + off; off += (size_t)MROWS * DINNER;
  float* xd = ws + off; off += (size_t)MROWS * XDCOLS;
  float* dt = ws + off; off += (size_t)MROWS * DINNER;
  float* y  = ws + off; off += (size_t)MROWS * DINNER;
  float* yg = ws + off; off += (size_t)MROWS * DINNER;
  float* Pb = ws + off; off += (size_t)NCHAINS * NCHUNK;
  float* Sb = ws + off; off += (size_t)NCHAINS * NCHUNK;
  float* h0 = ws + off; off += (size_t)NCHAINS * NCHUNK;

  inproj_kernel<<<((size_t)MROWS * DTRACK + 255) / 256, 256, 0, stream>>>(tracks, ipW, ipB, x0);

  const int scanWaves  = SEQ * (DINNER / 2) * NCHUNK;  // 16384
  const int scanBlocks = scanWaves / 8;                // 2048

  float* xcur = x0;
  for (int layer = 0; layer < 2; ++layer) {
    const int base = 3 + 19 * layer;
    const float* norm_scale = (const float*)d_in[base + 0];
    const float* lin_W      = (const float*)d_in[base + 1];
    const float* lin_b      = (const float*)d_in[base + 2];
    const float* lout_W     = (const float*)d_in[base + 17];
    const float* lout_b     = (const float*)d_in[base + 18];

    rmsnorm_kernel<<<MROWS / 8, 256, 0, stream>>>(xcur, norm_scale, xn);
    // xz = xn @ in_W + in_b : M=32768, K=64, N=256  (fp32 WMMA, LDS-staged B)
    gemm_wmma_f32<DTRACK, 4><<<dim3((2 * DINNER) / 64, MROWS / 128), 256, 0, stream>>>(
        xn, lin_W, lin_b, nullptr, xz, 2 * DINNER);

    for (int dir = 0; dir < 2; ++dir) {
      const int db = base + 3 + 7 * dir;
      const float* convW  = (const float*)d_in[db + 0];
      const float* convB  = (const float*)d_in[db + 1];
      const float* xprojW = (const float*)d_in[db + 2];
      const float* dtW    = (const float*)d_in[db + 3];
      const float* dtB    = (const float*)d_in[db + 4];
      const float* Alog   = (const float*)d_in[db + 5];
      const float* Dp     = (const float*)d_in[db + 6];

      conv_silu_kernel<<<((size_t)MROWS * DINNER + 255) / 256, 256, 0, stream>>>(
          xz, convW, convB, u, dir);
      xproj_kernel<<<((size_t)MROWS * XDCOLS + 255) / 256, 256, 0, stream>>>(u, xprojW, xd);
      dtproj_kernel<<<((size_t)MROWS * DINNER + 255) / 256, 256, 0, stream>>>(xd, dtW, dtB, dt);

      scan_pass1_kernel<<<scanBlocks, 256, 0, stream>>>(dt, u, xd, Alog, Pb, Sb);
      scan_combine_kernel<<<(NCHAINS + 255) / 256, 256, 0, stream>>>(Pb, Sb, h0);
      scan_pass3_kernel<<<scanBlocks, 256, 0, stream>>>(
          dt, u, xd, Alog, Dp, h0, y, /*reverse=*/dir, /*accumulate=*/dir);
    }

    gate_kernel<<<((size_t)MROWS * DINNER + 255) / 256, 256, 0, stream>>>(y, xz, yg);
    // out = yg @ out_W + out_b + residual : M=32768, K=128, N=64 (fp32 WMMA)
    float* xnext = (layer == 0) ? x1 : (float*)d_out;
    gemm_wmma_f32<DINNER, 4><<<dim3(DTRACK / 64, MROWS / 128), 256, 0, stream>>>(
        yg, lout_W, lout_b, xcur, xnext, DTRACK);
    xcur = xnext;
  }
}